// ModuleMamba_89318139887970
// MI455X (gfx1250) — compile-verified
//
#include <hip/hip_runtime.h>
#include <cmath>

#define B__ 4
#define L__ 2048
#define D_IN 128
#define D_MODEL 256
#define N_LAYERS 2
#define D_INNER 512
#define D_STATE 16
#define D_CONV 4
#define DT_RANK 16
#define M_TOT (B__ * L__) /* 8192 rows */
#define DBL_W (DT_RANK + 2 * D_STATE) /* 48 */
#define SC_CHUNKS 16
#define SC_LC (L__ / SC_CHUNKS) /* 128 steps per chunk */

typedef float v2f __attribute__((ext_vector_type(2)));
typedef float v8f __attribute__((ext_vector_type(8)));

__device__ __forceinline__ float silu_f(float x) { return x / (1.0f + __expf(-x)); }
__device__ __forceinline__ float softplus_f(float x) { return (x > 20.0f) ? x : log1pf(__expf(x)); }

enum { EPI_STORE = 0, EPI_BIAS = 1, EPI_BIAS_SOFTPLUS = 2, EPI_ACCUM = 3 };

// C[m,n] = sum_k A[m,k] * W[n,k]  (+bias / softplus / +=).
// One wave computes a 16 x (16*NT) strip with V_WMMA_F32_16X16X4_F32.
// All dims are compile-time: K-loop fully unrolls -> loads become
// base+imm-offset b64s the scheduler can clause ahead of the wmmas; an
// explicit 2-stage pipeline prefetches k+4 fragments before k's wmmas.
// Layouts per ISA 7.12.2: A lane holds row (lane&15), K-pair (lane>>4)*2;
// B mirrors over N; C/D: N = lane&15, M = vgpr + 8*(lane>>4).
template <int EPI, int NT, int K, int LDA, int LDW, int LDC>
__global__ __launch_bounds__(256) void wmma_gemm_nt(
    const float* __restrict__ A, const float* __restrict__ W,
    const float* __restrict__ bias, float* __restrict__ C) {
  const int lane = threadIdx.x & 31;
  const int wave = threadIdx.x >> 5;
  const int mtile = blockIdx.y * 8 + wave;
  const int ntile0 = blockIdx.x * NT;
  const int r = lane & 15;
  const int kh = (lane >> 4) << 1; // 0 or 2: which K pair this half-wave holds
  const float* Arow = A + (size_t)(mtile * 16 + r) * LDA + kh;
  const float* Wr[NT];
#pragma unroll
  for (int j = 0; j < NT; ++j)
    Wr[j] = W + (size_t)((ntile0 + j) * 16 + r) * LDW + kh;

  v8f acc[NT];
#pragma unroll
  for (int j = 0; j < NT; ++j) acc[j] = v8f{0.f, 0.f, 0.f, 0.f, 0.f, 0.f, 0.f, 0.f};

  // stage-0 preload
  v2f a_cur = *(const v2f*)(Arow);
  v2f b_cur[NT];
#pragma unroll
  for (int j = 0; j < NT; ++j) b_cur[j] = *(const v2f*)(Wr[j]);

#pragma unroll
  for (int k0 = 0; k0 < K; k0 += 4) {
    v2f a_nxt = {0.f, 0.f};
    v2f b_nxt[NT];
#pragma unroll
    for (int j = 0; j < NT; ++j) b_nxt[j] = v2f{0.f, 0.f};
    if (k0 + 4 < K) { // prefetch next fragments before consuming current ones
      a_nxt = *(const v2f*)(Arow + k0 + 4);
#pragma unroll
      for (int j = 0; j < NT; ++j) b_nxt[j] = *(const v2f*)(Wr[j] + k0 + 4);
    }
#pragma unroll
    for (int j = 0; j < NT; ++j)
      acc[j] = __builtin_amdgcn_wmma_f32_16x16x4_f32(
          /*neg_a=*/false, a_cur, /*neg_b=*/false, b_cur[j],
          /*c_mod=*/(short)0, acc[j], /*reuse_a=*/false, /*reuse_b=*/false);
    a_cur = a_nxt;
#pragma unroll
    for (int j = 0; j < NT; ++j) b_cur[j] = b_nxt[j];
  }

  const int mbase = mtile * 16 + (lane >> 4) * 8;
#pragma unroll
  for (int j = 0; j < NT; ++j) {
    const int n = (ntile0 + j) * 16 + r;
    float bv = (EPI == EPI_BIAS || EPI == EPI_BIAS_SOFTPLUS) ? bias[n] : 0.0f;
#pragma unroll
    for (int v = 0; v < 8; ++v) {
      float val = acc[j][v];
      if (EPI == EPI_BIAS) val += bv;
      if (EPI == EPI_BIAS_SOFTPLUS) val = softplus_f(val + bv);
      float* p = C + (size_t)(mbase + v) * LDC + n;
      if (EPI == EPI_ACCUM) *p += val; else *p = val;
    }
  }
}

// RMSNorm over D_MODEL=256: one wave per row, 8 elems/lane, wave32 shuffle reduce.
__global__ __launch_bounds__(256) void rmsnorm_kernel(
    const float* __restrict__ h, const float* __restrict__ w, float* __restrict__ out) {
  const int lane = threadIdx.x & 31;
  const int wave = threadIdx.x >> 5;
  const int row = blockIdx.x * 8 + wave;
  const float* hr = h + (size_t)row * D_MODEL;
  float vals[8];
  float ss = 0.f;
#pragma unroll
  for (int i = 0; i < 8; ++i) {
    float v = hr[lane + 32 * i];
    vals[i] = v;
    ss += v * v;
  }
#pragma unroll
  for (int off = 16; off >= 1; off >>= 1) ss += __shfl_xor(ss, off, 32);
  const float rs = rsqrtf(ss * (1.0f / (float)D_MODEL) + 1e-5f);
  float* orow = out + (size_t)row * D_MODEL;
#pragma unroll
  for (int i = 0; i < 8; ++i) {
    int e = lane + 32 * i;
    orow[e] = vals[i] * rs * w[e];
  }
}

// Causal depthwise conv (taps=4) + bias + SiLU. Reads xb half of xz (row stride 1024).
__global__ __launch_bounds__(256) void conv_silu_kernel(
    const float* __restrict__ xz, const float* __restrict__ cw,
    const float* __restrict__ cb, float* __restrict__ out) {
  const int idx = blockIdx.x * 256 + threadIdx.x; // (b*L + l)*512 + e
  const int e = idx & (D_INNER - 1);
  const int bl = idx >> 9;
  const int l = bl & (L__ - 1);
  const int b = bl >> 11;
  float acc = cb[e];
#pragma unroll
  for (int j = 0; j < D_CONV; ++j) {
    int t = l + j - (D_CONV - 1);
    if (t >= 0)
      acc += cw[e * D_CONV + j] * xz[(size_t)(b * L__ + t) * (2 * D_INNER) + e];
  }
  out[idx] = silu_f(acc);
}

// ---- Chunked selective scan (3 passes) ----
// h_t = exp(dt*A_n) h_{t-1} + dt*u*B_t ; chunk decay = exp(A_n * sum(dt))
// since A_n is constant -> chunk aggregate = {sum_dt, h_end[16]}.
// Pass1: per (b,e,chunk) run chunk from h=0; store sum_dt + h_end (32768 thr).
// Pass2: per (b,e) sequentially compose 16 chunk aggregates -> h_init per chunk.
// Pass3: per (b,e,chunk) rerun chunk from h_init, emit y gated with silu(z).

__global__ __launch_bounds__(256) void scan_pass1(
    const float* __restrict__ xbc, const float* __restrict__ delta,
    const float* __restrict__ dbl, const float* __restrict__ A_log,
    float* __restrict__ ssum, float* __restrict__ hend) {
  const int tid = blockIdx.x * 256 + threadIdx.x; // b*CHUNKS*512 range
  const int e = tid & (D_INNER - 1);
  const int bc_ = tid >> 9; // b*CHUNKS + chunk
  const int chunk = bc_ & (SC_CHUNKS - 1);
  const int b = bc_ >> 4;
  float Arow[D_STATE], hs[D_STATE];
#pragma unroll
  for (int n = 0; n < D_STATE; ++n) {
    Arow[n] = -__expf(A_log[e * D_STATE + n]);
    hs[n] = 0.f;
  }
  float S = 0.f;
  const int t0 = chunk * SC_LC;
  for (int t = t0; t < t0 + SC_LC; ++t) {
    const size_t rowE = (size_t)(b * L__ + t) * D_INNER + e;
    const float dt = delta[rowE];
    const float du = dt * xbc[rowE];
    const float* bc = dbl + (size_t)(b * L__ + t) * DBL_W + DT_RANK;
    S += dt;
#pragma unroll
    for (int n = 0; n < D_STATE; ++n)
      hs[n] = __expf(dt * Arow[n]) * hs[n] + du * bc[n];
  }
  ssum[(size_t)bc_ * D_INNER + e] = S;
#pragma unroll
  for (int n = 0; n < D_STATE; ++n)
    hend[((size_t)bc_ * D_STATE + n) * D_INNER + e] = hs[n];
}

__global__ __launch_bounds__(256) void scan_pass2(
    const float* __restrict__ A_log, const float* __restrict__ ssum,
    const float* __restrict__ hend, float* __restrict__ hinit) {
  const int tid = blockIdx.x * 256 + threadIdx.x; // 0 .. B*D_INNER-1
  const int e = tid & (D_INNER - 1);
  const int b = tid >> 9;
  float Arow[D_STATE], H[D_STATE];
#pragma unroll
  for (int n = 0; n < D_STATE; ++n) {
    Arow[n] = -__expf(A_log[e * D_STATE + n]);
    H[n] = 0.f;
  }
  for (int c = 0; c < SC_CHUNKS; ++c) {
    const int bc_ = b * SC_CHUNKS + c;
#pragma unroll
    for (int n = 0; n < D_STATE; ++n)
      hinit[((size_t)bc_ * D_STATE + n) * D_INNER + e] = H[n];
    const float S = ssum[(size_t)bc_ * D_INNER + e];
#pragma unroll
    for (int n = 0; n < D_STATE; ++n)
      H[n] = __expf(S * Arow[n]) * H[n] + hend[((size_t)bc_ * D_STATE + n) * D_INNER + e];
  }
}

__global__ __launch_bounds__(256) void scan_pass3(
    const float* __restrict__ xbc, const float* __restrict__ delta,
    const float* __restrict__ dbl, const float* __restrict__ xz,
    const float* __restrict__ A_log, const float* __restrict__ Dp,
    const float* __restrict__ hinit, float* __restrict__ y) {
  const int tid = blockIdx.x * 256 + threadIdx.x;
  const int e = tid & (D_INNER - 1);
  const int bc_ = tid >> 9;
  const int chunk = bc_ & (SC_CHUNKS - 1);
  const int b = bc_ >> 4;
  float Arow[D_STATE], hs[D_STATE];
#pragma unroll
  for (int n = 0; n < D_STATE; ++n) {
    Arow[n] = -__expf(A_log[e * D_STATE + n]);
    hs[n] = hinit[((size_t)bc_ * D_STATE + n) * D_INNER + e];
  }
  const float dpe = Dp[e];
  const int t0 = chunk * SC_LC;
  for (int t = t0; t < t0 + SC_LC; ++t) {
    const size_t rowE = (size_t)(b * L__ + t) * D_INNER + e;
    const float dt = delta[rowE];
    const float u = xbc[rowE];
    const float du = dt * u;
    const float* bc = dbl + (size_t)(b * L__ + t) * DBL_W + DT_RANK;
    float yv = 0.f;
#pragma unroll
    for (int n = 0; n < D_STATE; ++n) {
      hs[n] = __expf(dt * Arow[n]) * hs[n] + du * bc[n];
      yv += hs[n] * bc[D_STATE + n];
    }
    const float zv = xz[(size_t)(b * L__ + t) * (2 * D_INNER) + D_INNER + e];
    y[rowE] = (yv + u * dpe) * silu_f(zv);
  }
}

extern "C" void kernel_launch(void* const* d_in, const int* in_sizes, int n_in,
                              void* d_out, int out_size, void* d_ws, size_t ws_size,
                              hipStream_t stream) {
  const float* x      = (const float*)d_in[0];
  const float* W1     = (const float*)d_in[1];
  const float* b1     = (const float*)d_in[2];
  const float* W2     = (const float*)d_in[3];
  const float* b2     = (const float*)d_in[4];
  const float* norm_w = (const float*)d_in[5];
  const float* in_w   = (const float*)d_in[6];
  const float* conv_w = (const float*)d_in[7];
  const float* conv_b = (const float*)d_in[8];
  const float* xp_w   = (const float*)d_in[9];
  const float* dt_w   = (const float*)d_in[10];
  const float* dt_b   = (const float*)d_in[11];
  const float* A_log  = (const float*)d_in[12];
  const float* D_par  = (const float*)d_in[13];
  const float* out_w  = (const float*)d_in[14];

  float* ws = (float*)d_ws;
  float* h     = ws; ws += (size_t)M_TOT * D_MODEL;      // 8 MB
  float* un    = ws; ws += (size_t)M_TOT * D_MODEL;      // 8 MB
  float* xz    = ws; ws += (size_t)M_TOT * 2 * D_INNER;  // 32 MB (xb | z)
  float* xbc   = ws; ws += (size_t)M_TOT * D_INNER;      // 16 MB
  float* dblb  = ws; ws += (size_t)M_TOT * DBL_W;        // 1.5 MB
  float* delta = ws; ws += (size_t)M_TOT * D_INNER;      // 16 MB
  float* yb    = ws; ws += (size_t)M_TOT * D_INNER;      // 16 MB
  float* ssum  = ws; ws += (size_t)B__ * SC_CHUNKS * D_INNER;            // 128 KB
  float* hend  = ws; ws += (size_t)B__ * SC_CHUNKS * D_STATE * D_INNER;  // 2 MB
  float* hinit = ws; ws += (size_t)B__ * SC_CHUNKS * D_STATE * D_INNER;  // 2 MB

  const dim3 blk(256);
  const int mty = M_TOT / 128; // 64: 8 M-tiles (waves) per block
  const int sc_grid = (B__ * SC_CHUNKS * D_INNER) / 256; // 128 blocks

  // h = x @ W1^T + b1    (M=8192, N=256, K=128)
  wmma_gemm_nt<EPI_BIAS, 4, D_IN, D_IN, D_IN, D_MODEL>
      <<<dim3(D_MODEL / 64, mty), blk, 0, stream>>>(x, W1, b1, h);

  for (int l = 0; l < N_LAYERS; ++l) {
    rmsnorm_kernel<<<M_TOT / 8, blk, 0, stream>>>(h, norm_w + (size_t)l * D_MODEL, un);
    // xz = un @ in_w^T   (N=1024, K=256)
    wmma_gemm_nt<EPI_STORE, 4, D_MODEL, D_MODEL, D_MODEL, 2 * D_INNER>
        <<<dim3(2 * D_INNER / 64, mty), blk, 0, stream>>>(
            un, in_w + (size_t)l * 2 * D_INNER * D_MODEL, nullptr, xz);
    conv_silu_kernel<<<(M_TOT * D_INNER) / 256, blk, 0, stream>>>(
        xz, conv_w + (size_t)l * D_INNER * D_CONV, conv_b + (size_t)l * D_INNER, xbc);
    // dbl = xbc @ xp_w^T (N=48, K=512) -> NT=3
    wmma_gemm_nt<EPI_STORE, 3, D_INNER, D_INNER, D_INNER, DBL_W>
        <<<dim3(1, mty), blk, 0, stream>>>(
            xbc, xp_w + (size_t)l * DBL_W * D_INNER, nullptr, dblb);
    // delta = softplus(dbl[:, :16] @ dt_w^T + dt_b)  (N=512, K=16, lda=48)
    wmma_gemm_nt<EPI_BIAS_SOFTPLUS, 4, DT_RANK, DBL_W, DT_RANK, D_INNER>
        <<<dim3(D_INNER / 64, mty), blk, 0, stream>>>(
            dblb, dt_w + (size_t)l * D_INNER * DT_RANK, dt_b + (size_t)l * D_INNER, delta);
    // chunked selective scan, gated output into yb
    const float* Al = A_log + (size_t)l * D_INNER * D_STATE;
    scan_pass1<<<sc_grid, blk, 0, stream>>>(xbc, delta, dblb, Al, ssum, hend);
    scan_pass2<<<(B__ * D_INNER) / 256, blk, 0, stream>>>(Al, ssum, hend, hinit);
    scan_pass3<<<sc_grid, blk, 0, stream>>>(xbc, delta, dblb, xz, Al,
                                            D_par + (size_t)l * D_INNER, hinit, yb);
    // h += yb @ out_w^T  (N=256, K=512) — residual
    wmma_gemm_nt<EPI_ACCUM, 4, D_INNER, D_INNER, D_INNER, D_MODEL>
        <<<dim3(D_MODEL / 64, mty), blk, 0, stream>>>(
            yb, out_w + (size_t)l * D_MODEL * D_INNER, nullptr, h);
  }

  // out = h @ W2^T + b2  (N=128, K=256)
  wmma_gemm_nt<EPI_BIAS, 4, D_MODEL, D_MODEL, D_MODEL, D_IN>
      <<<dim3(D_IN / 64, mty), blk, 0, stream>>>(h, W2, b2, (float*)d_out);
}